// AttentionNodeLabelAggregation_5153960755613
// MI455X (gfx1250) — compile-verified
//
#include <hip/hip_runtime.h>
#include <hip/hip_bf16.h>

#define N_NODES 50000
#define N_EDGES 640000
#define D 128

typedef __attribute__((ext_vector_type(2))) float v2f;
typedef __attribute__((ext_vector_type(8))) float v8f;

// ---------- helpers: order-preserving float <-> uint key for atomic max ----------
__device__ __forceinline__ unsigned fkey(float f) {
    unsigned u = __float_as_uint(f);
    return (u & 0x80000000u) ? ~u : (u | 0x80000000u);
}
__device__ __forceinline__ float fkey_inv(unsigned k) {
    unsigned u = (k & 0x80000000u) ? (k & 0x7FFFFFFFu) : ~k;
    return __uint_as_float(u);
}

// ---------- 1) init: out[:, :D] = x, out[:, D:] = 0, segkey = 0, denom = 0 ----------
__global__ void init_out_kernel(const float* __restrict__ x, float* __restrict__ out,
                                unsigned* __restrict__ segkey, float* __restrict__ denom) {
    int t = blockIdx.x * blockDim.x + threadIdx.x;      // one thread per float4 of out
    if (t >= N_NODES * 64) return;                      // 2*D/4 = 64 float4 per row
    int node = t >> 6;
    int q = t & 63;
    float4* orow = (float4*)(out + (size_t)node * (2 * D));
    if (q < 32) {
        orow[q] = ((const float4*)(x + (size_t)node * D))[q];
    } else {
        orow[q] = make_float4(0.f, 0.f, 0.f, 0.f);
    }
    if (q == 0) {
        segkey[node] = 0u;          // <= key(any finite float); empty nodes never read back
        denom[node] = 0.f;
    }
}

// ---------- 2) gate GEMV via V_WMMA_F32_16X16X4_F32 ----------
// One wave32 computes gate for 16 nodes. A = x[base:base+16, k:k+4] (16x4 f32),
// B[k][n] = w[k] for all n (so every C column equals the dot product).
// A layout (ISA 7.12.2, 32-bit A 16x4): lanes 0-15 hold M=lane, VGPR0=K0, VGPR1=K1;
// lanes 16-31 hold M=lane-16, VGPR0=K2, VGPR1=K3. B 4x16 mirrors (K striped the same way).
// C layout: VGPR r, lanes 0-15 -> C[r][lane]; lanes 16-31 -> C[8+r][lane-16].
__global__ void gate_wmma_kernel(const float* __restrict__ x, const float* __restrict__ w,
                                 const float* __restrict__ b, float* __restrict__ gate) {
    int wave = (blockIdx.x * blockDim.x + threadIdx.x) >> 5;
    int lane = threadIdx.x & 31;
    int base = wave * 16;
    if (base >= N_NODES) return;                 // wave-uniform: EXEC stays all-1s below

    int row   = base + (lane & 15);              // node this lane's A element belongs to
    int khalf = (lane >> 4) * 2;                 // 0 for lanes 0-15, 2 for lanes 16-31

    v8f c = {};
    #pragma unroll 4
    for (int kc = 0; kc < D / 4; ++kc) {
        int col = kc * 4 + khalf;
        v2f a, bm;
        a.x  = x[(size_t)row * D + col];
        a.y  = x[(size_t)row * D + col + 1];
        bm.x = w[col];
        bm.y = w[col + 1];
        // 8 args: (neg_a, A, neg_b, B, c_mod, C, reuse_a, reuse_b)
        c = __builtin_amdgcn_wmma_f32_16x16x4_f32(false, a, false, bm,
                                                  (short)0, c, false, false);
    }
    float bias = b[0];
    // lane 0 owns C[0..7][0] -> nodes base..base+7; lane 16 owns C[8..15][0] -> base+8..base+15
    if (lane == 0 || lane == 16) {
        int m0 = base + ((lane >> 4) << 3);
        #pragma unroll
        for (int r = 0; r < 8; ++r) {
            gate[m0 + r] = c[r] + bias;
        }
    }
}

// ---------- 3) per-edge segment max over dst (atomic max on monotone uint key) ----------
__global__ void edge_max_kernel(const long long* __restrict__ ei, const float* __restrict__ gate,
                                unsigned* __restrict__ segkey) {
    int e = blockIdx.x * blockDim.x + threadIdx.x;
    if (e >= N_EDGES) return;
    int s = (int)ei[e];
    int d = (int)ei[N_EDGES + e];
    atomicMax(&segkey[d], fkey(gate[s]));
}

// ---------- 4) per-edge exp + segment sum (denominator) ----------
__global__ void edge_exp_kernel(const long long* __restrict__ ei, const float* __restrict__ gate,
                                const unsigned* __restrict__ segkey, float* __restrict__ escore,
                                float* __restrict__ denom) {
    int e = blockIdx.x * blockDim.x + threadIdx.x;
    if (e >= N_EDGES) return;
    int s = (int)ei[e];
    int d = (int)ei[N_EDGES + e];
    float m = fkey_inv(segkey[d]);
    unsigned mu = __float_as_uint(m);
    if ((mu & 0x7F800000u) == 0x7F800000u) m = 0.f;   // reference: non-finite seg_max -> 0
    float ee = __expf(gate[s] - m);
    escore[e] = ee;
    atomicAdd(&denom[d], ee);
}

// ---------- 5) weighted scatter-add: one wave32 per edge, float4 per lane ----------
__global__ void edge_scatter_kernel(const long long* __restrict__ ei, const float* __restrict__ x,
                                    const float* __restrict__ escore, const float* __restrict__ denom,
                                    float* __restrict__ out) {
    int wid = (blockIdx.x * blockDim.x + threadIdx.x) >> 5;
    int lane = threadIdx.x & 31;
    if (wid >= N_EDGES) return;                   // wave-uniform
    int s = (int)ei[wid];
    int d = (int)ei[N_EDGES + wid];
    float alpha = escore[wid] / fmaxf(denom[d], 1e-16f);
    float4 v = ((const float4*)(x + (size_t)s * D))[lane];   // coalesced 512B row gather (L2-hot)
    float* o = out + (size_t)d * (2 * D) + D + lane * 4;
    atomicAdd(o + 0, alpha * v.x);
    atomicAdd(o + 1, alpha * v.y);
    atomicAdd(o + 2, alpha * v.z);
    atomicAdd(o + 3, alpha * v.w);
}

extern "C" void kernel_launch(void* const* d_in, const int* in_sizes, int n_in,
                              void* d_out, int out_size, void* d_ws, size_t ws_size,
                              hipStream_t stream) {
    const float*     x  = (const float*)d_in[0];       // [N, D]
    const float*     gw = (const float*)d_in[1];       // [1, D] -> 128 floats
    const float*     gb = (const float*)d_in[2];       // [1]
    const long long* ei = (const long long*)d_in[3];   // [2, E] int64
    float* out = (float*)d_out;                        // [N, 2D]

    // workspace layout (256B aligned slots): gate[N] f32 | segkey[N] u32 | denom[N] f32 | escore[E] f32
    char* ws = (char*)d_ws;
    const size_t NA = ((size_t)N_NODES * 4 + 255) & ~(size_t)255;
    float*    gate   = (float*)(ws);
    unsigned* segkey = (unsigned*)(ws + NA);
    float*    denom  = (float*)(ws + 2 * NA);
    float*    escore = (float*)(ws + 3 * NA);

    const int BT = 256;

    // 1) concat-copy x into out[:, :D], zero agg half, reset reduction state
    init_out_kernel<<<(N_NODES * 64 + BT - 1) / BT, BT, 0, stream>>>(x, out, segkey, denom);

    // 2) gate GEMV on the matrix pipe (N/16 = 3125 waves, 32 WMMAs each)
    int gate_waves = N_NODES / 16;                       // 50000 % 16 == 0
    int gate_blocks = (gate_waves + (BT / 32) - 1) / (BT / 32);
    gate_wmma_kernel<<<gate_blocks, BT, 0, stream>>>(x, gw, gb, gate);

    // 3) scatter max
    edge_max_kernel<<<(N_EDGES + BT - 1) / BT, BT, 0, stream>>>(ei, gate, segkey);

    // 4) exp + denominator
    edge_exp_kernel<<<(N_EDGES + BT - 1) / BT, BT, 0, stream>>>(ei, gate, segkey, escore, denom);

    // 5) weighted feature scatter (wave per edge)
    long long scatter_threads = (long long)N_EDGES * 32;
    edge_scatter_kernel<<<(int)((scatter_threads + BT - 1) / BT), BT, 0, stream>>>(ei, x, escore, denom, out);
}